// LightBiHyperFormerConv_45621142618206
// MI455X (gfx1250) — compile-verified
//
#include <hip/hip_runtime.h>

typedef float v2f __attribute__((ext_vector_type(2)));
typedef float v8f __attribute__((ext_vector_type(8)));
typedef int   i32x4v __attribute__((vector_size(16)));

typedef __attribute__((address_space(1))) int*    gas_i32p;
typedef __attribute__((address_space(3))) int*    las_i32p;
typedef __attribute__((address_space(1))) i32x4v* gas_i128p;
typedef __attribute__((address_space(3))) i32x4v* las_i128p;

#define HN   4
#define DIN  257
#define DS   256
#define NR   20000
#define MRF  64

// CDNA5 async global->LDS staging (ASYNCcnt path), with synchronous fallback.
#if defined(__gfx1250__) &&                                            \
    __has_builtin(__builtin_amdgcn_global_load_async_to_lds_b32) &&    \
    __has_builtin(__builtin_amdgcn_global_load_async_to_lds_b128) &&   \
    __has_builtin(__builtin_amdgcn_s_wait_asynccnt)
#define HAS_ASYNC_LDS 1
#else
#define HAS_ASYNC_LDS 0
#endif

static __device__ __forceinline__ v8f wmma_f32(v2f a, v2f b, v8f c) {
  return __builtin_amdgcn_wmma_f32_16x16x4_f32(false, a, false, b, (short)0, c,
                                               false, false);
}

static __device__ __forceinline__ void a_copy_b32(float* l, const float* g) {
#if HAS_ASYNC_LDS
  __builtin_amdgcn_global_load_async_to_lds_b32(
      (gas_i32p)(void*)g, (las_i32p)(void*)l, 0, 0);
#else
  *l = *g;
#endif
}
static __device__ __forceinline__ void a_copy_b128(float* l, const float* g) {
#if HAS_ASYNC_LDS
  __builtin_amdgcn_global_load_async_to_lds_b128(
      (gas_i128p)(void*)g, (las_i128p)(void*)l, 0, 0);
#else
  *(float4*)l = *(const float4*)g;
#endif
}
static __device__ __forceinline__ void a_wait() {
#if HAS_ASYNC_LDS
  __builtin_amdgcn_s_wait_asynccnt(0);
#endif
}

// ---------------------------------------------------------------------------
// Kernel A: S = X*W[h] + b  (32 rows x 256 cols per block), fused epilogue:
//   type v  : val[h,n,:] = [sqrt(1+|S|^2) | S]
//   type k/q: feat[h,n,:] = exp((S*proj)/sqrt(tau)) * (exp(1-t^2)/16 + 1e-8)
// gridDim = (NR/32, HN, 6)  z: 0..2 = u{k,q,v}, 3..5 = i{k,q,v}
// Wave tiling: wave = (row group of 16) x (4 col tiles of 16) -> 4 wmma / A-frag
// ---------------------------------------------------------------------------
__global__ __launch_bounds__(256)
void hyp_proj_kernel(const float* __restrict__ Xu, const float* __restrict__ Xi,
                     const float* __restrict__ Wk, const float* __restrict__ Wq,
                     const float* __restrict__ Wv,
                     const float* __restrict__ bk, const float* __restrict__ bq,
                     const float* __restrict__ bv,
                     const float* __restrict__ proj,
                     float* __restrict__ val_u, float* __restrict__ val_i,
                     float* __restrict__ qp_u, float* __restrict__ kp_u,
                     float* __restrict__ qp_i, float* __restrict__ kp_i)
{
  __shared__ __align__(16) float ldsW[16 * 256]; // W k-chunk; reused as proj [64][64]
  __shared__ __align__(16) float ldsX[32 * 16];  // X k-chunk
  __shared__ __align__(16) float ldsS[32 * 256]; // S tile
  __shared__ float ldsSum[32];
  __shared__ float ldsT[32];

  const int tid  = threadIdx.x;
  const int lane = tid & 31;
  const int wid  = tid >> 5;
  const int hi   = lane >> 4;
  const int l16  = lane & 15;

  const int n0   = blockIdx.x * 32;
  const int h    = blockIdx.y;
  const int z    = blockIdx.z;
  const int isel = z / 3;   // 0=u 1=i
  const int type = z % 3;   // 0=k 1=q 2=v

  const float* X    = isel ? Xi : Xu;
  const float* W    = (type == 0) ? Wk : (type == 1) ? Wq : Wv;
  const float* bias = (type == 0) ? bk : (type == 1) ? bq : bv;
  const float* Wh   = W + (size_t)h * DIN * DS;

  if (tid < 32) ldsSum[tid] = 0.0f;

  const int rbase = (wid >> 2) * 16;   // wave row group: 0 or 16
  const int c0    = (wid & 3) * 64;    // wave col base: 4 tiles of 16

  v8f acc[4];
  const v8f vz = {0,0,0,0,0,0,0,0};
#pragma unroll
  for (int j = 0; j < 4; ++j) acc[j] = vz;

  for (int kc = 0; kc < 256; kc += 16) {          // full chunks, no guards
    for (int t = tid; t < 32 * 16; t += 256) {    // X: 32 rows x 16 k (4B align)
      int r = t >> 4, k = t & 15;
      a_copy_b32(&ldsX[t], X + (size_t)(n0 + r) * DIN + kc + k);
    }
    for (int t = tid; t < 16 * 64; t += 256) {    // W: 16 k x 256 cols (16B align)
      int kr = t >> 6, c4 = t & 63;
      a_copy_b128(&ldsW[kr * 256 + c4 * 4], Wh + (size_t)(kc + kr) * DS + c4 * 4);
    }
    a_wait();
    __syncthreads();
#pragma unroll
    for (int kk = 0; kk < 16; kk += 4) {
      v2f a;
      a.x = ldsX[(rbase + l16) * 16 + kk + 2 * hi];
      a.y = ldsX[(rbase + l16) * 16 + kk + 2 * hi + 1];
#pragma unroll
      for (int j = 0; j < 4; ++j) {
        v2f b;
        b.x = ldsW[(kk + 2 * hi) * 256 + c0 + 16 * j + l16];
        b.y = ldsW[(kk + 2 * hi + 1) * 256 + c0 + 16 * j + l16];
        acc[j] = wmma_f32(a, b, acc[j]);
      }
    }
    __syncthreads();
  }
  { // remainder chunk: only k = 256 is valid (guarded synchronous path)
    for (int t = tid; t < 32 * 16; t += 256) {
      int r = t >> 4, k = t & 15;
      ldsX[t] = (k == 0) ? X[(size_t)(n0 + r) * DIN + 256] : 0.0f;
    }
    for (int t = tid; t < 16 * 64; t += 256) {
      int kr = t >> 6, c4 = t & 63;
      float4 w4 = make_float4(0.f, 0.f, 0.f, 0.f);
      if (kr == 0) w4 = ((const float4*)(Wh + (size_t)256 * DS))[c4];
      ((float4*)ldsW)[t] = w4;
    }
    __syncthreads();
    v2f a;
    a.x = ldsX[(rbase + l16) * 16 + 2 * hi];
    a.y = ldsX[(rbase + l16) * 16 + 2 * hi + 1];
#pragma unroll
    for (int j = 0; j < 4; ++j) {
      v2f b;
      b.x = ldsW[(2 * hi) * 256 + c0 + 16 * j + l16];
      b.y = ldsW[(2 * hi + 1) * 256 + c0 + 16 * j + l16];
      acc[j] = wmma_f32(a, b, acc[j]);
    }
    __syncthreads();
  }

  // bias add, per-row sum of squares (shuffle reduce), stash S in LDS
  float bv4[4];
#pragma unroll
  for (int j = 0; j < 4; ++j) bv4[j] = bias[(size_t)h * DS + c0 + 16 * j + l16];
#pragma unroll
  for (int g = 0; g < 8; ++g) {
    int row = rbase + g + 8 * hi;
    float p = 0.0f;
#pragma unroll
    for (int j = 0; j < 4; ++j) {
      acc[j][g] += bv4[j];
      p += acc[j][g] * acc[j][g];
      ldsS[row * 256 + c0 + 16 * j + l16] = acc[j][g];
    }
#pragma unroll
    for (int m = 1; m < 16; m <<= 1) p += __shfl_xor(p, m, 16);
    if (l16 == 0) atomicAdd(&ldsSum[row], p);
  }
  __syncthreads();
  if (tid < 32) {
    float s = ldsSum[tid];
    // t^2 = 1 + s  ->  front = exp(1 - t^2)/16 + eps = exp(-s)*0.0625 + 1e-8
    ldsT[tid] = (type == 2) ? sqrtf(1.0f + s) : (expf(-s) * 0.0625f + 1e-8f);
  }
  __syncthreads();

  if (type == 2) {   // value: write [time | space]
    float* val = isel ? val_i : val_u;
    for (int t = tid; t < 32 * 256; t += 256) {
      int r = t >> 8, c = t & 255;
      val[((size_t)h * NR + n0 + r) * DIN + 1 + c] = ldsS[t];
    }
    if (tid < 32) val[((size_t)h * NR + n0 + tid) * DIN] = ldsT[tid];
    return;
  }

  // stage 2: F = exp((S * proj) * rsqrt(tau)) * front   (32 x 64, 8 tiles/8 waves)
  float* F = (type == 0) ? (isel ? kp_i : kp_u) : (isel ? qp_i : qp_u);
  const int f0 = (wid & 3) * 16;
  v8f fa = vz;
  for (int kc2 = 0; kc2 < 256; kc2 += 64) {
    for (int t = tid; t < 64 * 16; t += 256) {   // proj chunk 64 k x 64 cols
      int kr = t >> 4, c4 = t & 15;
      a_copy_b128(&ldsW[kr * MRF + c4 * 4], proj + (size_t)(kc2 + kr) * MRF + c4 * 4);
    }
    a_wait();
    __syncthreads();
#pragma unroll
    for (int kk = 0; kk < 64; kk += 4) {
      v2f a;
      a.x = ldsS[(rbase + l16) * 256 + kc2 + kk + 2 * hi];
      a.y = ldsS[(rbase + l16) * 256 + kc2 + kk + 2 * hi + 1];
      v2f b;
      b.x = ldsW[(kk + 2 * hi) * MRF + f0 + l16];
      b.y = ldsW[(kk + 2 * hi + 1) * MRF + f0 + l16];
      fa = wmma_f32(a, b, fa);
    }
    __syncthreads();
  }
#pragma unroll
  for (int g = 0; g < 8; ++g) {
    int row = rbase + g + 8 * hi;
    float wx = expf(fa[g] * 1.41421356237f);    // 1/sqrt(tau), tau=0.5
    F[((size_t)h * NR + n0 + row) * MRF + f0 + l16] = wx * ldsT[row];
  }
}

// ---------------------------------------------------------------------------
// Kernel B: last[h] (64x257) = sum_n kp[h,n,:]^T (outer) val[h,n,:]
// gridDim = (40, HN, 2);  dir 0: last_u = kp_i^T val_i ; dir 1: last_i = kp_u^T val_u
// ---------------------------------------------------------------------------
__global__ __launch_bounds__(256)
void last_kernel(const float* __restrict__ kp_u, const float* __restrict__ kp_i,
                 const float* __restrict__ val_u, const float* __restrict__ val_i,
                 float* __restrict__ last_u, float* __restrict__ last_i)
{
  __shared__ __align__(16) float ldsK[32 * 64];   // [k(n)][m]
  __shared__ __align__(16) float ldsV[32 * 272];  // [k(n)][d], zero-padded

  const int tid = threadIdx.x, lane = tid & 31, wid = tid >> 5;
  const int hi = lane >> 4, l16 = lane & 15;
  const int h = blockIdx.y, dir = blockIdx.z;

  const float* kp    = dir ? kp_u  : kp_i;
  const float* val   = dir ? val_u : val_i;
  float*       lastp = dir ? last_i : last_u;

  const int mt    = wid & 3;
  const int half  = wid >> 2;
  const int ntile = half ? 8 : 9;   // waves 0-3: N-tiles 0..8, waves 4-7: 9..16

  // pad columns d >= 257 stay zero for the whole kernel
  for (int t = tid; t < 32 * 272; t += 256) {
    int d = t % 272;
    if (d >= DIN) ldsV[t] = 0.0f;
  }

  v8f acc[9];
  const v8f vz = {0,0,0,0,0,0,0,0};
#pragma unroll
  for (int j = 0; j < 9; ++j) acc[j] = vz;

  for (int n0 = blockIdx.x * 32; n0 < NR; n0 += gridDim.x * 32) {
    for (int t = tid; t < 32 * 16; t += 256) {   // kp rows: 64 f32, 16B aligned
      int r = t >> 4, c4 = t & 15;
      a_copy_b128(&ldsK[r * 64 + c4 * 4],
                  kp + ((size_t)h * NR + n0 + r) * MRF + c4 * 4);
    }
    {
      const float* vb = val + ((size_t)h * NR + n0) * DIN;  // 32*257 contiguous
      for (int t = tid; t < 32 * DIN; t += 256) {
        int r = t / DIN, d = t - r * DIN;
        a_copy_b32(&ldsV[r * 272 + d], vb + t);
      }
    }
    a_wait();
    __syncthreads();
#pragma unroll
    for (int kk = 0; kk < 32; kk += 4) {
      v2f a;   // A[m][k] = kp[n0+k][m]
      a.x = ldsK[(kk + 2 * hi) * 64 + mt * 16 + l16];
      a.y = ldsK[(kk + 2 * hi + 1) * 64 + mt * 16 + l16];
#pragma unroll
      for (int j = 0; j < 9; ++j) {
        if (j < ntile) {
          int nt = half ? (9 + j) : j;
          v2f b;
          b.x = ldsV[(kk + 2 * hi) * 272 + nt * 16 + l16];
          b.y = ldsV[(kk + 2 * hi + 1) * 272 + nt * 16 + l16];
          acc[j] = wmma_f32(a, b, acc[j]);
        }
      }
    }
    __syncthreads();
  }

#pragma unroll
  for (int j = 0; j < 9; ++j) {
    if (j < ntile) {
      int nt = half ? (9 + j) : j;
#pragma unroll
      for (int g = 0; g < 8; ++g) {
        int m = mt * 16 + g + 8 * hi;
        int d = nt * 16 + l16;
        if (d < DIN)
          atomicAdd(&lastp[((size_t)h * MRF + m) * DIN + d], acc[j][g]);
      }
    }
  }
}

// ---------------------------------------------------------------------------
// Kernel C: z = qp*last, coeff = rsqrt(|mnorm(z)+eps|), mobius(val, z*coeff),
//           mean over heads, hyperbolic renorm, store.
// gridDim = (NR/16, 2)
// ---------------------------------------------------------------------------
__global__ __launch_bounds__(256)
void finalize_kernel(const float* __restrict__ qp_u, const float* __restrict__ qp_i,
                     const float* __restrict__ val_u, const float* __restrict__ val_i,
                     const float* __restrict__ last_u, const float* __restrict__ last_i,
                     float* __restrict__ out)
{
  __shared__ __align__(16) float ldsQ[16 * 64];
  __shared__ float ldsZ[16 * 272];
  __shared__ float ldsA[16 * 272];
  __shared__ float ldsAcc[16 * 257];
  __shared__ float ldsT[16];

  const int tid = threadIdx.x, lane = tid & 31, wid = tid >> 5;
  const int hi = lane >> 4, l16 = lane & 15;
  const int dir = blockIdx.y;
  const int n0  = blockIdx.x * 16;

  const float* qp    = dir ? qp_i  : qp_u;
  const float* val   = dir ? val_i : val_u;
  const float* lastp = dir ? last_i : last_u;
  float* outp = out + (size_t)dir * NR * DIN;

  for (int t = tid; t < 16 * 257; t += 256) ldsAcc[t] = 0.0f;
  for (int t = tid; t < 16 * 272; t += 256) {   // pad cols stay zero
    int d = t % 272;
    if (d >= DIN) ldsA[t] = 0.0f;
  }

  for (int h = 0; h < HN; ++h) {
    for (int t = tid; t < 16 * 16; t += 256) {   // qp rows: 64 f32, 16B aligned
      int r = t >> 4, c4 = t & 15;
      a_copy_b128(&ldsQ[r * 64 + c4 * 4],
                  qp + ((size_t)h * NR + n0 + r) * MRF + c4 * 4);
    }
    {
      const float* vb = val + ((size_t)h * NR + n0) * DIN;
      for (int t = tid; t < 16 * DIN; t += 256) {
        int r = t / DIN, d = t - r * DIN;
        a_copy_b32(&ldsA[r * 272 + d], vb + t);
      }
    }
    a_wait();
    __syncthreads();

    const float* lh = lastp + (size_t)h * MRF * DIN;
    for (int nt = wid; nt < 17; nt += 8) {
      v8f zv = {0,0,0,0,0,0,0,0};
      const int d = nt * 16 + l16;
#pragma unroll
      for (int kk = 0; kk < MRF; kk += 4) {
        v2f a;
        a.x = ldsQ[l16 * 64 + kk + 2 * hi];
        a.y = ldsQ[l16 * 64 + kk + 2 * hi + 1];
        v2f b;
        b.x = (d < DIN) ? lh[(size_t)(kk + 2 * hi) * DIN + d] : 0.0f;
        b.y = (d < DIN) ? lh[(size_t)(kk + 2 * hi + 1) * DIN + d] : 0.0f;
        zv = wmma_f32(a, b, zv);
      }
#pragma unroll
      for (int g = 0; g < 8; ++g)
        ldsZ[(g + 8 * hi) * 272 + nt * 16 + l16] = zv[g];
    }
    __syncthreads();

    if (tid < 16) {
      const int r = tid;
      float z0 = ldsZ[r * 272];
      float ssq = 0.0f;
      for (int d = 1; d < DIN; ++d) { float zz = ldsZ[r * 272 + d]; ssq += zz * zz; }
      float mn = -z0 * z0 + ssq;                       // mnorm(z)
      float coeff = rsqrtf(fabsf(mn + 1e-7f));         // 1/sqrt(|mnorm+eps|)
      float ab = 0.f, a2 = 0.f, b2 = 0.f;
      for (int d = 0; d < DIN; ++d) {
        float av = ldsA[r * 272 + d];
        float bv = ldsZ[r * 272 + d] * coeff;
        ab += av * bv; a2 += av * av; b2 += bv * bv;
      }
      float ca  = 1.0f + 2.0f * ab + b2;               // c = 1
      float cb  = 1.0f - a2;
      float inv = 1.0f / (1.0f + 2.0f * ab + a2 * b2 + 1e-7f);
      for (int d = 0; d < DIN; ++d) {
        float av = ldsA[r * 272 + d];
        float bv = ldsZ[r * 272 + d] * coeff;
        ldsAcc[r * 257 + d] += (ca * av + cb * bv) * inv;
      }
    }
    __syncthreads();
  }

  if (tid < 16) {
    const int r = tid;
    float m0 = ldsAcc[r * 257] * 0.25f;
    float ssq = 0.0f;
    for (int d = 1; d < DIN; ++d) { float mv = ldsAcc[r * 257 + d] * 0.25f; ssq += mv * mv; }
    float mn = -m0 * m0 + ssq;
    ldsT[r] = 0.25f * rsqrtf(fabsf(mn) + 1e-7f);       // fold mean /4 + renorm
  }
  __syncthreads();
  for (int t = tid; t < 16 * 257; t += 256) {
    int r = t / 257, d = t - r * 257;
    outp[(size_t)(n0 + r) * DIN + d] = ldsAcc[r * 257 + d] * ldsT[r];
  }
}

// ---------------------------------------------------------------------------
extern "C" void kernel_launch(void* const* d_in, const int* in_sizes, int n_in,
                              void* d_out, int out_size, void* d_ws, size_t ws_size,
                              hipStream_t stream)
{
  const float* u    = (const float*)d_in[0];
  const float* iin  = (const float*)d_in[1];
  const float* Wk   = (const float*)d_in[2];
  const float* bk   = (const float*)d_in[3];
  const float* Wq   = (const float*)d_in[4];
  const float* bq   = (const float*)d_in[5];
  const float* Wv   = (const float*)d_in[6];
  const float* bv   = (const float*)d_in[7];
  const float* proj = (const float*)d_in[8];
  float* out = (float*)d_out;
  float* ws  = (float*)d_ws;

  const size_t VAL  = (size_t)HN * NR * DIN;   // 20,560,000 floats
  const size_t FEAT = (size_t)HN * NR * MRF;   //  5,120,000 floats
  const size_t LAST = (size_t)HN * MRF * DIN;  //     65,792 floats
  float* val_u  = ws;
  float* val_i  = val_u + VAL;
  float* qp_u   = val_i + VAL;
  float* kp_u   = qp_u + FEAT;
  float* qp_i   = kp_u + FEAT;
  float* kp_i   = qp_i + FEAT;
  float* last_u = kp_i + FEAT;
  float* last_i = last_u + LAST;
  // total ~247 MB of workspace

  (void)hipMemsetAsync(last_u, 0, 2 * LAST * sizeof(float), stream);

  hyp_proj_kernel<<<dim3(NR / 32, HN, 6), 256, 0, stream>>>(
      u, iin, Wk, Wq, Wv, bk, bq, bv, proj,
      val_u, val_i, qp_u, kp_u, qp_i, kp_i);

  last_kernel<<<dim3(40, HN, 2), 256, 0, stream>>>(
      kp_u, kp_i, val_u, val_i, last_u, last_i);

  finalize_kernel<<<dim3(NR / 16, 2), 256, 0, stream>>>(
      qp_u, qp_i, val_u, val_i, last_u, last_i, out);
}